// NGCF_4982162063609
// MI455X (gfx1250) — compile-verified
//
#include <hip/hip_runtime.h>

// ---------------------------------------------------------------------------
// NGCF forward for MI455X (gfx1250, wave32).
//   cur  : [N_NODES,64] f32 embedding buffer (in-place updated per layer)
//   nb   : [N_NODES,64] f32 scatter buffer for SpMM
// Per layer: memset(nb,0) -> spmm scatter (fp32 atomics, L2-resident) ->
//            fused WMMA row-transform (x@W1+b1, nb@W2+b2, (nbl*x)@W2+b2,
//            leaky-relu) -> gather batch rows into output slice.
// FP32 WMMA (v_wmma_f32_16x16x4_f32) keeps full fp32 precision vs reference.
// Weights are staged in LDS in a row-pair-interleaved layout so every
// B-fragment is a single ds_load_b64.
// ---------------------------------------------------------------------------

#define NGCF_N_USERS 100000
#define NGCF_N_ITEMS 50000
#define NGCF_N_NODES 150000
#define NGCF_N_EDGES 4800000
#define NGCF_EMB 64
#define NGCF_BATCH 16384
#define NGCF_TILES ((NGCF_N_NODES + 15) / 16)   // 9375 (exact)

typedef __attribute__((ext_vector_type(2))) float v2f;
typedef __attribute__((ext_vector_type(8))) float v8f;

// ---------------------------------------------------------------------------
// Copy user_emb ++ item_emb into cur, float4 granularity.
// ---------------------------------------------------------------------------
__global__ void ngcf_init_emb(const float* __restrict__ ue,
                              const float* __restrict__ ie,
                              float* __restrict__ cur) {
    long i = (long)blockIdx.x * blockDim.x + threadIdx.x;   // float4 index
    const long nU4 = (long)NGCF_N_USERS * (NGCF_EMB / 4);
    const long nT4 = (long)NGCF_N_NODES * (NGCF_EMB / 4);
    if (i >= nT4) return;
    float4 v = (i < nU4) ? ((const float4*)ue)[i]
                         : ((const float4*)ie)[i - nU4];
    ((float4*)cur)[i] = v;
}

// ---------------------------------------------------------------------------
// SpMM scatter: 16 lanes per edge (2 edges per wave); float4 per lane.
// Gather source row from cur (L2-resident), scale by val, fp32 atomic-add
// into nb[row]. global_atomic_add_f32; both buffers fit in the 192MB L2.
// ---------------------------------------------------------------------------
__global__ __launch_bounds__(256)
void ngcf_spmm_scatter(const int* __restrict__ rows,
                       const int* __restrict__ cols,
                       const float* __restrict__ vals,
                       const float* __restrict__ cur,
                       float* __restrict__ nb) {
    unsigned t = blockIdx.x * 256u + threadIdx.x;
    unsigned edge = t >> 4;              // 16 lanes per edge
    int sub = threadIdx.x & 15;
    if (edge >= NGCF_N_EDGES) return;
    int r = rows[edge];
    int c = cols[edge];
    float v = vals[edge];
    float4 x = ((const float4*)(cur + (size_t)c * NGCF_EMB))[sub];
    float* dst = nb + (size_t)r * NGCF_EMB + sub * 4;
    atomicAdd(dst + 0, v * x.x);
    atomicAdd(dst + 1, v * x.y);
    atomicAdd(dst + 2, v * x.z);
    atomicAdd(dst + 3, v * x.w);
}

// ---------------------------------------------------------------------------
// One 16x64 @ 64x64 GEMM accumulate with V_WMMA_F32_16X16X4_F32.
// A fragment (16x4 f32, ISA 7.12.2): row m = lane&15, K = 4k + v + 2*(lane>>4)
//   -> contiguous pair in row-major A => one ds_load_b64.
// B fragment: Wp is row-pair interleaved (Wp[p*128 + n*2 + (r&1)] = W[r][n],
//   p = r>>1) so the (v=0,v=1) pair for K rows (4k+2h, 4k+2h+1) is one b64.
// D layout: M = vgpr + 8*(lane>>4), N = lane&15.
// If MUL, A element = A[idx] * A2[idx] (interact term).
// ---------------------------------------------------------------------------
template <bool MUL>
__device__ inline void ngcf_gemm64(const float* __restrict__ A,
                                   const float* __restrict__ A2,
                                   const float* __restrict__ Wp,
                                   v8f acc[4], int lane) {
    const int mh = lane & 15;
    const int half = lane >> 4;
#pragma unroll
    for (int k = 0; k < 16; ++k) {
        const int k0 = 4 * k + 2 * half;     // even K row
        v2f a = *(const v2f*)(A + mh * NGCF_EMB + k0);
        if (MUL) {
            v2f a2 = *(const v2f*)(A2 + mh * NGCF_EMB + k0);
            a.x *= a2.x;
            a.y *= a2.y;
        }
        const int p = k0 >> 1;               // row-pair index = 2k + half
#pragma unroll
        for (int j = 0; j < 4; ++j) {
            v2f b = *(const v2f*)(Wp + p * 128 + (j * 16 + mh) * 2);
            acc[j] = __builtin_amdgcn_wmma_f32_16x16x4_f32(
                false, a, false, b, (short)0, acc[j], false, false);
        }
    }
}

// ---------------------------------------------------------------------------
// Fused per-layer row transform. Block = 256 threads = 8 waves; each wave
// owns one 16-row tile (block covers 128 rows). In-place update of cur.
//   nbl = nb@W2 + b2
//   out = leaky( x@W1 + b1 + nbl + (nbl*x)@W2 + b2 )
// ---------------------------------------------------------------------------
__global__ __launch_bounds__(256)
void ngcf_layer_update(float* __restrict__ cur,
                       const float* __restrict__ nb,
                       const float* __restrict__ W1,
                       const float* __restrict__ b1,
                       const float* __restrict__ W2,
                       const float* __restrict__ b2) {
    __shared__ float sW1[NGCF_EMB * NGCF_EMB];   // 16 KB, pair-interleaved
    __shared__ float sW2[NGCF_EMB * NGCF_EMB];   // 16 KB, pair-interleaved
    __shared__ float sB1[NGCF_EMB];
    __shared__ float sB2[NGCF_EMB];
    __shared__ float sX[8][16 * NGCF_EMB];       // 32 KB (per-wave X tiles)
    __shared__ float sN[8][16 * NGCF_EMB];       // 32 KB (nb raw, then nbl)

    const int tid = threadIdx.x;
    const int wave = tid >> 5;
    const int lane = tid & 31;
    const int mh = lane & 15;
    const int half = lane >> 4;

    // Stage weights pair-interleaved: sW[p*128 + n*2 + e] = W[2p+e][n].
    // Consecutive threads -> consecutive n -> coalesced global loads and
    // conflict-free stride-2-word ds_store_b64.
#pragma unroll
    for (int q = tid; q < 32 * NGCF_EMB; q += 256) {   // 2048 row-pairs*cols
        int p = q >> 6;
        int n = q & 63;
        float2 w1; w1.x = W1[(2 * p) * NGCF_EMB + n];
        w1.y = W1[(2 * p + 1) * NGCF_EMB + n];
        float2 w2; w2.x = W2[(2 * p) * NGCF_EMB + n];
        w2.y = W2[(2 * p + 1) * NGCF_EMB + n];
        ((float2*)sW1)[p * 64 + n] = w1;
        ((float2*)sW2)[p * 64 + n] = w2;
    }
    if (tid < NGCF_EMB) { sB1[tid] = b1[tid]; sB2[tid] = b2[tid]; }

    const long tile = (long)blockIdx.x * 8 + wave;
    const long row0 = tile * 16;
    const bool active = (row0 < NGCF_N_NODES);   // wave-uniform -> EXEC all 1s

    if (active) {
        const float4* gx = (const float4*)(cur + row0 * NGCF_EMB);
        const float4* gn = (const float4*)(nb + row0 * NGCF_EMB);
        float4* lx = (float4*)sX[wave];
        float4* ln = (float4*)sN[wave];
#pragma unroll
        for (int i = lane; i < 16 * NGCF_EMB / 4; i += 32) {
            lx[i] = gx[i];
            ln[i] = gn[i];
        }
    }
    __syncthreads();

    if (active) {
        // ---- nbl = nb @ W2 + b2 ----
        v8f accN[4] = {};
        ngcf_gemm64<false>(sN[wave], nullptr, sW2, accN, lane);
        // Relayout nbl (with bias) into sN row-major for the GEMM3
        // A-fragment pass (raw nb fully consumed above; same-wave LDS ops
        // are in-order). accN stays live for the final combine.
#pragma unroll
        for (int j = 0; j < 4; ++j)
#pragma unroll
            for (int g = 0; g < 8; ++g) {
                const int m = g + 8 * half;
                const int n = j * 16 + mh;
                sN[wave][m * NGCF_EMB + n] = accN[j][g] + sB2[n];
            }

        // ---- self = x @ W1 (bias added in combine) ----
        v8f accS[4] = {};
        ngcf_gemm64<false>(sX[wave], nullptr, sW1, accS, lane);

        // ---- interact = (nbl * x) @ W2 (bias added in combine) ----
        v8f accI[4] = {};
        ngcf_gemm64<true>(sN[wave], sX[wave], sW2, accI, lane);

        // ---- combine + leaky relu; overwrite sX (x fully consumed) ----
        // out = accS + b1 + (accN + b2) + (accI + b2)
#pragma unroll
        for (int j = 0; j < 4; ++j) {
            const int n = j * 16 + mh;
            const float bias = sB1[n] + 2.0f * sB2[n];
#pragma unroll
            for (int g = 0; g < 8; ++g) {
                const int m = g + 8 * half;
                float v = accS[j][g] + accN[j][g] + accI[j][g] + bias;
                v = (v > 0.0f) ? v : 0.2f * v;
                sX[wave][m * NGCF_EMB + n] = v;
            }
        }

        // Coalesced writeback (same-wave LDS in-order: stores precede loads).
        float4* gx = (float4*)(cur + row0 * NGCF_EMB);
        const float4* lx = (const float4*)sX[wave];
#pragma unroll
        for (int i = lane; i < 16 * NGCF_EMB / 4; i += 32) gx[i] = lx[i];
    }
}

// ---------------------------------------------------------------------------
// Gather current embeddings for (users, pos, neg) into output slice `s`:
// out[(t*BATCH + i)*256 + s*64 .. +64] = cur[node]. One wave per batch row.
// ---------------------------------------------------------------------------
__global__ __launch_bounds__(256)
void ngcf_gather_slice(const int* __restrict__ users,
                       const int* __restrict__ pos,
                       const int* __restrict__ neg,
                       const float* __restrict__ cur,
                       float* __restrict__ out, int slice) {
    int g = (int)((blockIdx.x * (unsigned)blockDim.x + threadIdx.x) >> 5);
    int lane = threadIdx.x & 31;
    if (g >= 3 * NGCF_BATCH) return;
    int t = g / NGCF_BATCH;
    int i = g - t * NGCF_BATCH;
    int node;
    if (t == 0)      node = users[i];
    else if (t == 1) node = NGCF_N_USERS + pos[i];
    else             node = NGCF_N_USERS + neg[i];
    const float2* src = (const float2*)(cur + (size_t)node * NGCF_EMB);
    float2 x = src[lane];
    float2* dst = (float2*)(out + (size_t)g * 256 + (size_t)slice * NGCF_EMB);
    dst[lane] = x;
}

// ---------------------------------------------------------------------------
extern "C" void kernel_launch(void* const* d_in, const int* in_sizes, int n_in,
                              void* d_out, int out_size, void* d_ws, size_t ws_size,
                              hipStream_t stream) {
    (void)in_sizes; (void)n_in; (void)out_size; (void)ws_size;

    const int*   users    = (const int*)d_in[0];
    const int*   pos      = (const int*)d_in[1];
    const int*   neg      = (const int*)d_in[2];
    const int*   rows     = (const int*)d_in[3];
    const int*   cols     = (const int*)d_in[4];
    const float* vals     = (const float*)d_in[5];
    const float* user_emb = (const float*)d_in[6];
    const float* item_emb = (const float*)d_in[7];
    const float* W1s      = (const float*)d_in[8];
    const float* b1s      = (const float*)d_in[9];
    const float* W2s      = (const float*)d_in[10];
    const float* b2s      = (const float*)d_in[11];

    float* cur = (float*)d_ws;                                  // 38.4 MB
    float* nb  = cur + (size_t)NGCF_N_NODES * NGCF_EMB;         // 38.4 MB
    float* out = (float*)d_out;

    // Layer-0 embeddings.
    {
        long n4 = (long)NGCF_N_NODES * (NGCF_EMB / 4);
        ngcf_init_emb<<<(unsigned)((n4 + 255) / 256), 256, 0, stream>>>(
            user_emb, item_emb, cur);
    }
    const unsigned gatherBlocks = (3 * NGCF_BATCH + 7) / 8;      // 8 waves/block
    ngcf_gather_slice<<<gatherBlocks, 256, 0, stream>>>(users, pos, neg, cur, out, 0);

    const unsigned spmmBlocks  = (NGCF_N_EDGES + 15) / 16;       // 16 edges/block
    const unsigned layerBlocks = (NGCF_TILES + 7) / 8;           // 1 wave/16-row tile

    for (int l = 0; l < 3; ++l) {
        hipMemsetAsync(nb, 0, (size_t)NGCF_N_NODES * NGCF_EMB * sizeof(float), stream);
        ngcf_spmm_scatter<<<spmmBlocks, 256, 0, stream>>>(rows, cols, vals, cur, nb);
        ngcf_layer_update<<<layerBlocks, 256, 0, stream>>>(
            cur, nb,
            W1s + (size_t)l * NGCF_EMB * NGCF_EMB, b1s + (size_t)l * NGCF_EMB,
            W2s + (size_t)l * NGCF_EMB * NGCF_EMB, b2s + (size_t)l * NGCF_EMB);
        ngcf_gather_slice<<<gatherBlocks, 256, 0, stream>>>(users, pos, neg, cur, out, l + 1);
    }
}